// SparseMultiHeadGraphAttention_75007308858099
// MI455X (gfx1250) — compile-verified
//
#include <hip/hip_runtime.h>
#include <hip/hip_bf16.h>

// ---------------------------------------------------------------------------
// Sparse multi-head GAT for MI455X (gfx1250, wave32).
// Key simplification: edge scores depend only on the SOURCE node (the
// reference uses src features for both score terms), so scores collapse to a
// per-node quantity. Everything except the 1.7M-edge scatter-add is cheap.
// GEMM runs on V_WMMA_F32_16X16X4_F32 to keep fp32 precision.
// ---------------------------------------------------------------------------

#define N_NODES 100000
#define N_EDGES 1600000
#define F_IN    64
#define N_HEAD  4
#define F_OUT   16
#define HF      (N_HEAD * F_OUT)   // 64

typedef __attribute__((ext_vector_type(2))) float v2f;
typedef __attribute__((ext_vector_type(8))) float v8f;

// ---- order-preserving float<->uint encoding for atomic max ----------------
__device__ __forceinline__ unsigned enc_f32(float f) {
    unsigned u = __float_as_uint(f);
    return (u & 0x80000000u) ? ~u : (u | 0x80000000u);
}
__device__ __forceinline__ float dec_f32(unsigned u) {
    return (u & 0x80000000u) ? __uint_as_float(u & 0x7fffffffu)
                             : __uint_as_float(~u);
}

// ---------------------------------------------------------------------------
// K1: h_prime = h @ W + b   (100000x64 @ 64x64, fp32 WMMA)
// Block = 128 threads = 4 waves; wave w computes the 16x16 tile at col w*16.
// Grid = N_NODES/16 = 6250 (N divisible by 16, no remainder handling).
// A layout (16x4 f32): lane<16 -> M=lane, v0=K0,v1=K1; lane>=16 -> K=2,3.
// B layout (4x16 f32): lane half selects K row pair, N = lane&15.
// C layout: VGPR i -> M=i (lanes 0-15) / M=i+8 (lanes 16-31), N=lane&15.
// ---------------------------------------------------------------------------
__global__ void __launch_bounds__(128)
gat_gemm_wmma(const float* __restrict__ h, const float* __restrict__ W,
              const float* __restrict__ bias, float* __restrict__ hp) {
    const int wave = threadIdx.x >> 5;
    const int lane = threadIdx.x & 31;
    const int row0 = blockIdx.x << 4;
    const int col0 = wave << 4;
    const int m    = lane & 15;          // row (A) / col (B,C)
    const int half = lane >> 4;          // 0 or 1
    const int kh   = half << 1;          // K sub-offset within 4-wide step

    v8f c = {};
    const float* arow = h + (row0 + m) * F_IN;
    #pragma unroll
    for (int k = 0; k < F_IN; k += 4) {
        v2f a;
        a.x = arow[k + kh];
        a.y = arow[k + kh + 1];
        v2f b;
        b.x = W[(k + kh) * HF + col0 + m];
        b.y = W[(k + kh + 1) * HF + col0 + m];
        c = __builtin_amdgcn_wmma_f32_16x16x4_f32(
                /*neg_a=*/false, a, /*neg_b=*/false, b,
                /*c_mod=*/(short)0, c, /*reuse_a=*/false, /*reuse_b=*/false);
    }

    const int   col = col0 + m;
    const float bv  = bias[col];
    #pragma unroll
    for (int i = 0; i < 8; ++i) {
        const int row = row0 + i + (half << 3);
        hp[row * HF + col] = c[i] + bv;
    }
}

// ---------------------------------------------------------------------------
// K2: per-node scores  s[n,h] = leaky_relu( dot16(h_prime[n,h,:], a_src+a_trg) )
// One thread per (n, head).
// ---------------------------------------------------------------------------
__global__ void gat_scores(const float* __restrict__ hp,
                           const float* __restrict__ a_src,
                           const float* __restrict__ a_trg,
                           float* __restrict__ s) {
    const int idx = blockIdx.x * blockDim.x + threadIdx.x;
    if (idx >= N_NODES * N_HEAD) return;
    const int n  = idx >> 2;
    const int hh = idx & 3;
    const float* v = hp + n * HF + hh * F_OUT;
    const float* a = a_src + hh * F_OUT;
    const float* b = a_trg + hh * F_OUT;
    float e = 0.f;
    #pragma unroll
    for (int f = 0; f < F_OUT; ++f) e += v[f] * (a[f] + b[f]);
    s[idx] = (e > 0.f) ? e : 0.2f * e;
}

// ---------------------------------------------------------------------------
// K3: global max of s (400K elems). Grid-stride, wave shfl reduce, atomicMax
// on order-preserving uint encoding (buffer pre-zeroed == -inf sentinel).
// ---------------------------------------------------------------------------
__global__ void gat_maxreduce(const float* __restrict__ s,
                              unsigned* __restrict__ mbuf) {
    const int total = N_NODES * N_HEAD;
    float m = -3.4e38f;
    for (int i = blockIdx.x * blockDim.x + threadIdx.x; i < total;
         i += gridDim.x * blockDim.x)
        m = fmaxf(m, s[i]);
    #pragma unroll
    for (int off = 16; off >= 1; off >>= 1)
        m = fmaxf(m, __shfl_xor(m, off, 32));
    if ((threadIdx.x & 31) == 0) atomicMax(mbuf, enc_f32(m));
}

// ---------------------------------------------------------------------------
// K4: p[n,h] = exp(s[n,h] - M)   (in place over s)
//     w[n,h,:] = p[n,h] * h_prime[n,h,:]  (in place over hp)
// One thread per (n, head); 16 contiguous floats scaled via float4.
// ---------------------------------------------------------------------------
__global__ void gat_expscale(float* __restrict__ sp, float* __restrict__ hp,
                             const unsigned* __restrict__ mbuf) {
    const int idx = blockIdx.x * blockDim.x + threadIdx.x;
    if (idx >= N_NODES * N_HEAD) return;
    const float M = dec_f32(*mbuf);
    const float p = __expf(sp[idx] - M);
    sp[idx] = p;
    float4* v = reinterpret_cast<float4*>(hp + idx * F_OUT);
    #pragma unroll
    for (int q = 0; q < 4; ++q) {
        float4 t = v[q];
        t.x *= p; t.y *= p; t.z *= p; t.w *= p;
        v[q] = t;
    }
}

// ---------------------------------------------------------------------------
// K5: edge scatter. 64 threads per edge (incl. the N self-loops):
//   out[h, trg, f] += w[src, h, f]        (fp32 global atomics, hits L2)
//   denom[trg, h]  += p[src, h]           (one lane per head)
// Accumulates directly into the transposed [H, N, F] output layout.
// ---------------------------------------------------------------------------
__global__ void gat_edge_scatter(const int* __restrict__ ei,
                                 const float* __restrict__ w,
                                 const float* __restrict__ p,
                                 float* __restrict__ out,
                                 float* __restrict__ denom) {
    const long long tid =
        (long long)blockIdx.x * blockDim.x + threadIdx.x;
    const long long total = (long long)(N_EDGES + N_NODES) * HF;
    if (tid >= total) return;
    const int e  = (int)(tid >> 6);
    const int j  = (int)(tid & 63);
    const int hh = j >> 4;
    const int f  = j & 15;
    int src, trg;
    if (e < N_EDGES) {
        src = ei[e];
        trg = ei[N_EDGES + e];
    } else {
        src = trg = e - N_EDGES;   // self-loop
    }
    atomicAdd(&out[(size_t)hh * (N_NODES * F_OUT) + trg * F_OUT + f],
              w[(size_t)src * HF + j]);
    if (f == 0)
        atomicAdd(&denom[trg * N_HEAD + hh], p[src * N_HEAD + hh]);
}

// ---------------------------------------------------------------------------
// K6: out[h,n,f] /= (denom[n,h] + 1e-10). Fully coalesced on out.
// ---------------------------------------------------------------------------
__global__ void gat_normalize(float* __restrict__ out,
                              const float* __restrict__ denom) {
    const int tid = blockIdx.x * blockDim.x + threadIdx.x;
    if (tid >= N_NODES * HF) return;
    const int hh = tid / (N_NODES * F_OUT);
    const int n  = (tid >> 4) % N_NODES;
    out[tid] = out[tid] / (denom[n * N_HEAD + hh] + 1e-10f);
}

// ---------------------------------------------------------------------------
extern "C" void kernel_launch(void* const* d_in, const int* in_sizes, int n_in,
                              void* d_out, int out_size, void* d_ws, size_t ws_size,
                              hipStream_t stream) {
    const float* h     = (const float*)d_in[0];
    const float* lin_w = (const float*)d_in[1];
    const float* lin_b = (const float*)d_in[2];
    const float* a_src = (const float*)d_in[3];
    const float* a_trg = (const float*)d_in[4];
    const int*   ei    = (const int*)d_in[5];
    float* out = (float*)d_out;

    // scratch layout (all 256B aligned):
    //   [0,           25.6MB) h_prime -> w (in place)
    //   [25.6MB,      27.2MB) s -> p (in place)
    //   [27.2MB,      28.8MB) denom
    //   [28.8MB,      +4B   ) encoded global max
    char* ws = (char*)d_ws;
    float*    hp    = (float*)(ws);
    float*    sp    = (float*)(ws + 25600000);
    float*    denom = (float*)(ws + 27200000);
    unsigned* mbuf  = (unsigned*)(ws + 28800000);

    hipMemsetAsync(out,   0, (size_t)N_NODES * HF * sizeof(float), stream);
    hipMemsetAsync(denom, 0, (size_t)N_NODES * N_HEAD * sizeof(float), stream);
    hipMemsetAsync(mbuf,  0, sizeof(unsigned), stream);

    // K1: WMMA GEMM, one 16-row strip per block, 4 waves cover 64 cols.
    gat_gemm_wmma<<<N_NODES / 16, 128, 0, stream>>>(h, lin_w, lin_b, hp);

    // K2: per-node leaky-relu scores.
    {
        int n = N_NODES * N_HEAD;
        gat_scores<<<(n + 255) / 256, 256, 0, stream>>>(hp, a_src, a_trg, sp);
    }

    // K3: global max.
    gat_maxreduce<<<1024, 256, 0, stream>>>(sp, mbuf);

    // K4: exp + scale in place.
    {
        int n = N_NODES * N_HEAD;
        gat_expscale<<<(n + 255) / 256, 256, 0, stream>>>(sp, hp, mbuf);
    }

    // K5: edge scatter-add (dominant cost; L2-resident gathers + fp32 atomics).
    {
        long long total = (long long)(N_EDGES + N_NODES) * HF;
        int blocks = (int)((total + 255) / 256);
        gat_edge_scatter<<<blocks, 256, 0, stream>>>(ei, hp, sp, out, denom);
    }

    // K6: normalize.
    {
        int n = N_NODES * HF;
        gat_normalize<<<(n + 255) / 256, 256, 0, stream>>>(out, denom);
    }
}